// GAT_3547642987042
// MI455X (gfx1250) — compile-verified
//
#include <hip/hip_runtime.h>
#include <hip/hip_bf16.h>
#include <math.h>

#define DHID 128
#define NHEAD 2
#define NEG_ATT 0.2f

typedef float v2f __attribute__((ext_vector_type(2)));
typedef float v8f __attribute__((ext_vector_type(8)));

// ---------------------------------------------------------------------------
// x = h @ W^T with full-precision V_WMMA_F32_16X16X4_F32.
// One wave -> one 16x16 output tile. Block = 256 threads = 8 waves covering
// the 8 column tiles of D=128 for one 16-row tile. N % 16 == 0 (50000/16=3125)
// so EXEC is all-ones for every WMMA (ISA requirement).
//
// fp32 A 16x4 layout: lane<16: M=lane, v0=K0,v1=K1 ; lane>=16: M=lane-16, K2,K3
// fp32 B 4x16 layout (rows striped across lanes): lane<16: N=lane, K0,K1 ;
//                                                 lane>=16: K2,K3
// Both map to a contiguous 8-byte (v2f) load from the row-major operands.
// ---------------------------------------------------------------------------
__global__ __launch_bounds__(256) void gat_gemm_nt_f32(
    const float* __restrict__ A,   // (N,128) row-major
    const float* __restrict__ W,   // (128,128) row-major (torch Linear: out,in)
    float* __restrict__ X, int N) {
  const int wave = threadIdx.x >> 5;
  const int lane = threadIdx.x & 31;
  const int half = lane >> 4;         // 0: K{0,1}, 1: K{2,3}
  const int l16  = lane & 15;
  const int row0 = blockIdx.x * 16;
  const int col0 = wave * 16;
  const float* arow = A + (size_t)(row0 + l16) * DHID + 2 * half;
  const float* brow = W + (size_t)(col0 + l16) * DHID + 2 * half;
  v8f acc = {0.f, 0.f, 0.f, 0.f, 0.f, 0.f, 0.f, 0.f};
#pragma unroll 4
  for (int kk = 0; kk < DHID; kk += 4) {
    v2f a = *(const v2f*)(arow + kk);
    v2f b = *(const v2f*)(brow + kk);
    acc = __builtin_amdgcn_wmma_f32_16x16x4_f32(
        false, a, false, b, (short)0, acc, false, false);
  }
  // C/D layout: VGPR v, lane<16 -> (M=v, N=lane); lane>=16 -> (M=v+8)
  float* xout = X + (size_t)(row0 + 8 * half) * DHID + col0 + l16;
#pragma unroll
  for (int v = 0; v < 8; ++v) xout[(size_t)v * DHID] = acc[v];
}

// a_dst[n,h] = dot(x[n,h*64: h*64+64], att[h, 0:64])
// a_src[n,h] = dot(x[n,h*64: h*64+64], att[h,64:128])
__global__ void gat_att_sums(const float* __restrict__ x,
                             const float* __restrict__ att,  // (H, 2k)=(2,128)
                             float* __restrict__ a_dst, float* __restrict__ a_src,
                             int N) {
  int t = blockIdx.x * blockDim.x + threadIdx.x;
  if (t >= N * NHEAD) return;
  int n = t >> 1, h = t & 1;
  const float* xr = x + (size_t)n * DHID + h * 64;
  const float* ad = att + h * DHID;
  const float* as = ad + 64;
  float sd = 0.f, ss = 0.f;
#pragma unroll 8
  for (int j = 0; j < 64; ++j) {
    float v = xr[j];
    sd += v * ad[j];
    ss += v * as[j];
  }
  a_dst[t] = sd;
  a_src[t] = ss;
}

__global__ void gat_init_ms(float* __restrict__ m, float* __restrict__ s, int N) {
  int t = blockIdx.x * blockDim.x + threadIdx.x;
  if (t >= N * NHEAD) return;
  m[t] = -INFINITY;
  s[t] = 0.f;
}

__global__ void gat_init_out_bias(float* __restrict__ out,
                                  const float* __restrict__ bias, int total) {
  int t = blockIdx.x * blockDim.x + threadIdx.x;
  if (t >= total) return;
  out[t] = bias[t & (DHID - 1)];
}

__device__ __forceinline__ void edge_sd(const int* ei, int E, int e, int& s, int& d) {
  if (e < E) { s = ei[e]; d = ei[E + e]; }
  else       { s = d = e - E; }            // appended self loops
}

// pass 1: alpha = leaky_relu(a_dst[dst]+a_src[src]); segment max over src
__global__ void gat_edge_max(const int* __restrict__ ei, int E, int N,
                             const float* __restrict__ a_dst,
                             const float* __restrict__ a_src,
                             float* __restrict__ m) {
  int e = blockIdx.x * blockDim.x + threadIdx.x;
  if (e >= E + N) return;
  int s, d;
  edge_sd(ei, E, e, s, d);
#pragma unroll
  for (int h = 0; h < NHEAD; ++h) {
    float a = a_dst[d * NHEAD + h] + a_src[s * NHEAD + h];
    a = (a > 0.f) ? a : NEG_ATT * a;
    atomicMax(&m[s * NHEAD + h], a);       // global_atomic_max_num_f32
  }
}

// pass 2: e = exp(alpha - m[src]); segment sum over src; stash e per edge
__global__ void gat_edge_exp(const int* __restrict__ ei, int E, int N,
                             const float* __restrict__ a_dst,
                             const float* __restrict__ a_src,
                             const float* __restrict__ m,
                             float* __restrict__ ssum,
                             float* __restrict__ e_val) {
  int e = blockIdx.x * blockDim.x + threadIdx.x;
  if (e >= E + N) return;
  int s, d;
  edge_sd(ei, E, e, s, d);
#pragma unroll
  for (int h = 0; h < NHEAD; ++h) {
    float a = a_dst[d * NHEAD + h] + a_src[s * NHEAD + h];
    a = (a > 0.f) ? a : NEG_ATT * a;
    float ev = expf(a - m[s * NHEAD + h]);
    e_val[(size_t)e * NHEAD + h] = ev;
    atomicAdd(&ssum[s * NHEAD + h], ev);
  }
}

// pass 3: out[dst] += x[src] * (e/(sum[src]+eps)); wave per edge, float4/lane
__global__ __launch_bounds__(256) void gat_edge_scatter(
    const int* __restrict__ ei, int E, int N,
    const float* __restrict__ x, const float* __restrict__ e_val,
    const float* __restrict__ ssum, float* __restrict__ out) {
  int gid = blockIdx.x * blockDim.x + threadIdx.x;
  int e = gid >> 5;
  int lane = gid & 31;
  if (e >= E + N) return;
  int s, d;
  edge_sd(ei, E, e, s, d);
  int h = lane >> 4;                       // lanes 0-15: head 0, 16-31: head 1
  float a = e_val[(size_t)e * NHEAD + h] / (ssum[s * NHEAD + h] + 1e-16f);
  const float4 xv = *(const float4*)(x + (size_t)s * DHID + lane * 4);
  float* o = out + (size_t)d * DHID + lane * 4;
  atomicAdd(o + 0, xv.x * a);
  atomicAdd(o + 1, xv.y * a);
  atomicAdd(o + 2, xv.z * a);
  atomicAdd(o + 3, xv.w * a);
}

__global__ void gat_relu_inplace(float* __restrict__ v, int total) {
  int t = blockIdx.x * blockDim.x + threadIdx.x;
  if (t >= total) return;
  float x = v[t];
  v[t] = x > 0.f ? x : 0.f;
}

// output = x_embed @ W_out^T + b_out ; ypred = argmax (written as float)
__global__ void gat_proj(const float* __restrict__ xe,
                         const float* __restrict__ W_out,  // (3,128)
                         const float* __restrict__ b_out,
                         float* __restrict__ output, float* __restrict__ ypred,
                         int N) {
  int n = blockIdx.x * blockDim.x + threadIdx.x;
  if (n >= N) return;
  const float* xr = xe + (size_t)n * DHID;
  float acc0 = b_out[0], acc1 = b_out[1], acc2 = b_out[2];
#pragma unroll 8
  for (int j = 0; j < DHID; ++j) {
    float v = xr[j];
    acc0 += v * W_out[j];
    acc1 += v * W_out[DHID + j];
    acc2 += v * W_out[2 * DHID + j];
  }
  output[(size_t)n * 3 + 0] = acc0;
  output[(size_t)n * 3 + 1] = acc1;
  output[(size_t)n * 3 + 2] = acc2;
  int arg = 0;
  float best = acc0;
  if (acc1 > best) { best = acc1; arg = 1; }
  if (acc2 > best) { best = acc2; arg = 2; }
  ypred[n] = (float)arg;
}

__global__ void gat_gather(const float* __restrict__ output,
                           const float* __restrict__ ypred,
                           const int* __restrict__ node_index,
                           float* __restrict__ node_out,
                           float* __restrict__ y_nodepred, int NI) {
  int i = blockIdx.x * blockDim.x + threadIdx.x;
  if (i >= NI) return;
  int idx = node_index[i];
  node_out[(size_t)i * 3 + 0] = output[(size_t)idx * 3 + 0];
  node_out[(size_t)i * 3 + 1] = output[(size_t)idx * 3 + 1];
  node_out[(size_t)i * 3 + 2] = output[(size_t)idx * 3 + 2];
  y_nodepred[i] = ypred[idx];
}

extern "C" void kernel_launch(void* const* d_in, const int* in_sizes, int n_in,
                              void* d_out, int out_size, void* d_ws, size_t ws_size,
                              hipStream_t stream) {
  const float* x          = (const float*)d_in[0];
  const int*   ei         = (const int*)d_in[1];   // (2,E) src rows then dst
  const int*   node_index = (const int*)d_in[3];
  const float* Wl[3]   = {(const float*)d_in[4], (const float*)d_in[7], (const float*)d_in[10]};
  const float* attl[3] = {(const float*)d_in[5], (const float*)d_in[8], (const float*)d_in[11]};
  const float* bl[3]   = {(const float*)d_in[6], (const float*)d_in[9], (const float*)d_in[12]};
  const float* W_out = (const float*)d_in[13];
  const float* b_out = (const float*)d_in[14];

  const int N  = in_sizes[0] / DHID;   // 50000
  const int E  = in_sizes[1] / 2;      // 800000
  const int NI = in_sizes[3];          // 1000
  const int Et = E + N;                // with self loops

  // workspace carve-out (floats): ~86 MB
  float* ws = (float*)d_ws;
  size_t o = 0;
  float* xbuf  = ws + o; o += (size_t)N * DHID;   // post-GEMM features
  float* buf0  = ws + o; o += (size_t)N * DHID;   // layer1 out / layer2 in
  float* buf1  = ws + o; o += (size_t)N * DHID;   // layer2 out / layer3 in
  float* a_dst = ws + o; o += (size_t)N * NHEAD;
  float* a_src = ws + o; o += (size_t)N * NHEAD;
  float* mbuf  = ws + o; o += (size_t)N * NHEAD;
  float* sbuf  = ws + o; o += (size_t)N * NHEAD;
  float* e_val = ws + o; o += (size_t)Et * NHEAD;
  float* outp  = ws + o; o += (size_t)N * 3;      // final logits

  // d_out layout: x_embed | node_output | ypred | y_nodepred
  float* xembed    = (float*)d_out;
  float* node_out  = xembed + (size_t)N * DHID;
  float* ypred     = node_out + (size_t)NI * 3;
  float* ynodepred = ypred + N;

  const int TB = 256;
  const int gNH   = (N * NHEAD + TB - 1) / TB;
  const int gND   = (N * DHID + TB - 1) / TB;
  const int gE    = (Et + TB - 1) / TB;
  const int gE32  = (int)(((size_t)Et * 32 + TB - 1) / TB);
  const int gGemm = N / 16;                        // N is a multiple of 16

  const float* hin = x;
  float* outbuf[3] = {buf0, buf1, xembed};
  for (int L = 0; L < 3; ++L) {
    gat_gemm_nt_f32<<<gGemm, TB, 0, stream>>>(hin, Wl[L], xbuf, N);
    gat_att_sums<<<gNH, TB, 0, stream>>>(xbuf, attl[L], a_dst, a_src, N);
    gat_init_ms<<<gNH, TB, 0, stream>>>(mbuf, sbuf, N);
    gat_init_out_bias<<<gND, TB, 0, stream>>>(outbuf[L], bl[L], N * DHID);
    gat_edge_max<<<gE, TB, 0, stream>>>(ei, E, N, a_dst, a_src, mbuf);
    gat_edge_exp<<<gE, TB, 0, stream>>>(ei, E, N, a_dst, a_src, mbuf, sbuf, e_val);
    gat_edge_scatter<<<gE32, TB, 0, stream>>>(ei, E, N, xbuf, e_val, sbuf, outbuf[L]);
    gat_relu_inplace<<<gND, TB, 0, stream>>>(outbuf[L], N * DHID);
    hin = outbuf[L];
    // inter-layer leaky_relu(0.3) is identity on relu output -> dropped (exact)
  }

  gat_proj<<<(N + TB - 1) / TB, TB, 0, stream>>>(xembed, W_out, b_out, outp, ypred, N);
  gat_gather<<<(NI + TB - 1) / TB, TB, 0, stream>>>(outp, ypred, node_index, node_out, ynodepred, NI);
}